// Top1SparseMoEFFN_6330781794938
// MI455X (gfx1250) — compile-verified
//
#include <hip/hip_runtime.h>
#include <hip/hip_bf16.h>

// Top-1 Switch MoE FFN for MI455X (gfx1250), fp32 via V_WMMA_F32_16X16X4_F32.
//
// B=8, T=4096 -> N=32768 tokens, D=512, F=2048, E=8.
// Pipeline: init counters -> gate+route (atomic compaction) -> aux loss ->
// per-expert tiled FFN using wave32 WMMA with fp32 A/B/C (precision-safe).
//
// Tiling (moe_ffn): each wave owns BOTH 16-row bands and a private column
// range, so every W1/W2 B-fragment is loaded exactly once per workgroup and
// reused for 2 WMMAs -> ~1.25-2 memory issues per wmma, halved L2 traffic.

typedef float v2f __attribute__((ext_vector_type(2)));
typedef float v8f __attribute__((ext_vector_type(8)));

#define N_TOK 32768
#define DIM   512
#define FFD   2048
#define NEXP  8

// ---------------------------------------------------------------- init ----
__global__ void moe_init(int* __restrict__ counts) {
    if (threadIdx.x < NEXP) counts[threadIdx.x] = 0;
}

// ---------------------------------------------------------------- gate ----
__global__ __launch_bounds__(256) void moe_gate(
    const float* __restrict__ x,       // [N, D]
    const float* __restrict__ gate_w,  // [D, E]
    const float* __restrict__ gate_b,  // [E]
    int*   __restrict__ counts,        // [E]
    int*   __restrict__ etok,          // [E, N]
    float* __restrict__ p1)            // [N]
{
    __shared__ float sgw[DIM * NEXP];
    __shared__ float sgb[NEXP];
    for (int i = threadIdx.x; i < DIM * NEXP; i += 256) sgw[i] = gate_w[i];
    if (threadIdx.x < NEXP) sgb[threadIdx.x] = gate_b[threadIdx.x];
    __syncthreads();

    const int t = blockIdx.x * 256 + threadIdx.x;

    float l[NEXP];
    #pragma unroll
    for (int e = 0; e < NEXP; ++e) l[e] = sgb[e];

    const float4* xr = (const float4*)(x + (size_t)t * DIM);
    for (int d4 = 0; d4 < DIM / 4; ++d4) {
        float4 xv = xr[d4];
        const float* gw = sgw + d4 * 4 * NEXP;
        #pragma unroll
        for (int e = 0; e < NEXP; ++e) {
            l[e] += xv.x * gw[e]
                  + xv.y * gw[NEXP + e]
                  + xv.z * gw[2 * NEXP + e]
                  + xv.w * gw[3 * NEXP + e];
        }
    }

    // argmax (first max, matching jnp.argmax) + softmax top-1 prob
    float m = l[0];
    int ei = 0;
    #pragma unroll
    for (int e = 1; e < NEXP; ++e) {
        if (l[e] > m) { m = l[e]; ei = e; }
    }
    float s = 0.0f;
    #pragma unroll
    for (int e = 0; e < NEXP; ++e) s += __expf(l[e] - m);
    float p = 1.0f / s;  // exp(lmax - lmax)/s

    int slot = atomicAdd(&counts[ei], 1);
    etok[ei * N_TOK + slot] = t;
    p1[t] = p;
}

// ----------------------------------------------------------------- aux ----
__global__ void moe_aux(const int* __restrict__ counts, float* __restrict__ out_aux) {
    if (threadIdx.x == 0) {
        float a = 0.0f;
        #pragma unroll
        for (int e = 0; e < NEXP; ++e) {
            float u = (float)counts[e] * (1.0f / (float)N_TOK);
            a += u * u;
        }
        out_aux[0] = a * (float)NEXP;
    }
}

// ----------------------------------------------------------------- ffn ----
// Block = 256 threads (8 wave32). Tile = 32 tokens of one expert's list.
// Out[32x512] in registers: wave w owns cols [w*64, w*64+64), both row bands
// -> 8 accumulators (4 col-tiles x 2 bands) of v8f per wave.
// Loop F in chunks of FC=128:
//   GEMM1: H[32x128] = relu(X[32x512] @ W1[:,fc:fc+128] + b1)
//          (wave w computes col tile w*16, both 16-row bands; B reused x2)
//   GEMM2: acc += H[32x128] @ W2[fc:fc+128, w*64 : w*64+64]
__global__ __launch_bounds__(256) void moe_ffn(
    const float* __restrict__ x,    // [N, D]
    const float* __restrict__ w1,   // [E, D, F]
    const float* __restrict__ b1,   // [E, F]
    const float* __restrict__ w2,   // [E, F, D]
    const float* __restrict__ b2,   // [E, D]
    const int*   __restrict__ counts,
    const int*   __restrict__ etok,
    const float* __restrict__ p1,
    float*       __restrict__ out)  // [N, D]
{
    constexpr int MT = 32;         // token tile
    constexpr int XS = DIM + 4;    // padded X stride (bank-conflict-free)
    constexpr int FC = 128;        // F chunk
    constexpr int HS = FC + 4;     // padded H stride (132*4 % 256 == 16)

    extern __shared__ float smem[];
    float* sX = smem;                  // MT * XS
    float* sH = sX + MT * XS;          // MT * HS
    float* sP = sH + MT * HS;          // MT
    int*   sT = (int*)(sP + MT);       // MT

    const int e    = blockIdx.y;
    const int cnt  = counts[e];
    const int base = blockIdx.x * MT;
    if (base >= cnt) return;           // uniform early exit

    const float* w1e = w1 + (size_t)e * DIM * FFD;
    const float* w2e = w2 + (size_t)e * FFD * DIM;
    const float* b1e = b1 + e * FFD;
    const float* b2e = b2 + e * DIM;

    // ---- gather X tile into LDS (8 threads per row, float4) ----
    {
        const int row = threadIdx.x >> 3;
        const int seg = threadIdx.x & 7;
        int tok = -1;
        if (base + row < cnt) tok = etok[e * N_TOK + base + row];
        if (seg == 0) {
            sT[row] = tok;
            sP[row] = (tok >= 0) ? p1[tok] : 0.0f;
        }
        float4* dst = (float4*)(sX + row * XS + seg * 64);
        if (tok >= 0) {
            const float4* src = (const float4*)(x + (size_t)tok * DIM + seg * 64);
            #pragma unroll
            for (int i = 0; i < 16; ++i) dst[i] = src[i];
        } else {
            float4 z = make_float4(0.f, 0.f, 0.f, 0.f);
            #pragma unroll
            for (int i = 0; i < 16; ++i) dst[i] = z;
        }
    }
    __syncthreads();

    const int w     = threadIdx.x >> 5;  // wave 0..7
    const int lane  = threadIdx.x & 31;
    const int lhalf = lane >> 4;         // 0: lanes 0-15, 1: lanes 16-31
    const int l15   = lane & 15;

    // acc[ct*2 + band]: col tile ct (16 cols), row band (16 rows)
    v8f acc[8];
    #pragma unroll
    for (int t = 0; t < 8; ++t) acc[t] = {};

    for (int fc = 0; fc < FFD; fc += FC) {
        // ---- GEMM1: wave computes H col tile w*16 (cols fc+w*16..+15),
        //      both row bands; B fragment shared across the two bands. ----
        v8f h0 = {};   // rows 0..15
        v8f h1 = {};   // rows 16..31
        const int c1 = fc + w * 16 + l15;     // W1 column for this lane
        const float* w1c = w1e + c1;
        // warm L2 for next chunk's W1 column stream
        if (fc + FC < FFD) __builtin_prefetch(w1e + c1 + FC, 0, 1);
        #pragma unroll 2
        for (int k = 0; k < DIM; k += 4) {
            // A frags (16x4 f32): lanes 0-15 hold K=k,k+1; lanes 16-31 K=k+2,k+3
            const float* xr0 = sX + l15 * XS + k + 2 * lhalf;
            const float* xr1 = sX + (16 + l15) * XS + k + 2 * lhalf;
            v2f a0; a0.x = xr0[0]; a0.y = xr0[1];
            v2f a1; a1.x = xr1[0]; a1.y = xr1[1];
            // B frag (4x16 f32): VGPR0 row K (+2 hi lanes), VGPR1 row K+1 (+2)
            const int kk = k + 2 * lhalf;
            v2f b;
            b.x = w1c[(size_t)kk * FFD];
            b.y = w1c[(size_t)(kk + 1) * FFD];
            h0 = __builtin_amdgcn_wmma_f32_16x16x4_f32(
                    false, a0, false, b, (short)0, h0, false, false);
            h1 = __builtin_amdgcn_wmma_f32_16x16x4_f32(
                    false, a1, false, b, (short)0, h1, false, false);
        }
        const float bias1 = b1e[c1];
        const int colh = w * 16 + l15;        // local H column

        __syncthreads();  // prior GEMM2 done reading sH
        #pragma unroll
        for (int r = 0; r < 8; ++r) {
            float hv0 = h0[r] + bias1;
            float hv1 = h1[r] + bias1;
            hv0 = hv0 > 0.0f ? hv0 : 0.0f;
            hv1 = hv1 > 0.0f ? hv1 : 0.0f;
            const int rl = r + 8 * lhalf;     // C layout: M=r / M=r+8
            sH[rl * HS + colh]        = hv0;
            sH[(16 + rl) * HS + colh] = hv1;
        }
        __syncthreads();

        // ---- GEMM2: acc += H[32x128] @ W2[fc:fc+128, w*64 : w*64+64] ----
        #pragma unroll 2
        for (int k = 0; k < FC; k += 4) {
            const float* hr0 = sH + l15 * HS + k + 2 * lhalf;
            const float* hr1 = sH + (16 + l15) * HS + k + 2 * lhalf;
            v2f a0; a0.x = hr0[0]; a0.y = hr0[1];
            v2f a1; a1.x = hr1[0]; a1.y = hr1[1];
            const int kk = fc + k + 2 * lhalf;
            const float* w2r = w2e + (size_t)kk * DIM + w * 64 + l15;
            #pragma unroll
            for (int ct = 0; ct < 4; ++ct) {
                v2f b;
                b.x = w2r[ct * 16];
                b.y = w2r[ct * 16 + DIM];
                acc[ct * 2 + 0] = __builtin_amdgcn_wmma_f32_16x16x4_f32(
                        false, a0, false, b, (short)0, acc[ct * 2 + 0], false, false);
                acc[ct * 2 + 1] = __builtin_amdgcn_wmma_f32_16x16x4_f32(
                        false, a1, false, b, (short)0, acc[ct * 2 + 1], false, false);
            }
        }
    }

    // ---- epilogue: out = p * (acc + b2), scattered to token rows ----
    #pragma unroll
    for (int ct = 0; ct < 4; ++ct) {
        const int col = w * 64 + ct * 16 + l15;
        const float bias2 = b2e[col];
        #pragma unroll
        for (int band = 0; band < 2; ++band) {
            #pragma unroll
            for (int r = 0; r < 8; ++r) {
                const int row = band * 16 + r + 8 * lhalf;
                const int tok = sT[row];
                if (tok >= 0) {
                    out[(size_t)tok * DIM + col] =
                        sP[row] * (acc[ct * 2 + band][r] + bias2);
                }
            }
        }
    }
}

// -------------------------------------------------------------- launch ----
extern "C" void kernel_launch(void* const* d_in, const int* in_sizes, int n_in,
                              void* d_out, int out_size, void* d_ws, size_t ws_size,
                              hipStream_t stream) {
    const float* x      = (const float*)d_in[0];
    const float* gate_w = (const float*)d_in[1];
    const float* gate_b = (const float*)d_in[2];
    const float* w1     = (const float*)d_in[3];
    const float* b1     = (const float*)d_in[4];
    const float* w2     = (const float*)d_in[5];
    const float* b2     = (const float*)d_in[6];
    float* out = (float*)d_out;

    // workspace layout
    int*   counts = (int*)d_ws;                    // [16] (E used)
    int*   etok   = counts + 16;                   // [E*N]
    float* p1     = (float*)(etok + NEXP * N_TOK); // [N]

    moe_init<<<1, 64, 0, stream>>>(counts);
    moe_gate<<<N_TOK / 256, 256, 0, stream>>>(x, gate_w, gate_b, counts, etok, p1);
    moe_aux<<<1, 64, 0, stream>>>(counts, out + (size_t)N_TOK * DIM);

    const size_t shmem = (size_t)(32 * (DIM + 4) + 32 * (128 + 4) + 32) * sizeof(float)
                       + 32 * sizeof(int);
    moe_ffn<<<dim3(N_TOK / 32, NEXP), 256, shmem, stream>>>(
        x, w1, b1, w2, b2, counts, etok, p1, out);
}